// DRGSTEncoder_83545703842334
// MI455X (gfx1250) — compile-verified
//
#include <hip/hip_runtime.h>

// ---------------------------------------------------------------------------
// 3-layer GCN (PyG GCNConv semantics) for gfx1250 / MI455X, wave32.
//   per layer: h = dropout(h); hw = h @ W; agg = scatter(dst) hw[src]*norm
//              + hw*dinv^2 + b; relu between layers.
// GEMM uses V_WMMA_F32_16X16X4_F32 (f32 in/out, matches reference precision).
// ---------------------------------------------------------------------------

#define GCN_D 128
#define GCN_ROWS_PER_WG 128   // 8 row-tiles of 16 per workgroup
#define GCN_ROW_TILES 8

typedef __attribute__((ext_vector_type(2))) float v2f;
typedef __attribute__((ext_vector_type(8))) float v8f;

__device__ __forceinline__ unsigned pcg_hash(unsigned v) {
    v = v * 747796405u + 2891336453u;
    v = ((v >> ((v >> 28u) + 4u)) ^ v) * 277803737u;
    return (v >> 22u) ^ v;
}

// ---------------------------------------------------------------------------
// Degree / normalization helpers
// ---------------------------------------------------------------------------
__global__ void gcn_zero_f32(float* __restrict__ p, int n) {
    int i = blockIdx.x * blockDim.x + threadIdx.x;
    if (i < n) p[i] = 0.0f;
}

__global__ void gcn_deg_count(const long long* __restrict__ dst,
                              float* __restrict__ deg, int e) {
    int i = blockIdx.x * blockDim.x + threadIdx.x;
    if (i < e) atomicAdd(&deg[(int)dst[i]], 1.0f);
}

__global__ void gcn_deg_finalize(float* __restrict__ deg, int n) {
    int i = blockIdx.x * blockDim.x + threadIdx.x;
    if (i < n) deg[i] = rsqrtf(deg[i] + 1.0f);   // deg includes self-loop
}

// ---------------------------------------------------------------------------
// WMMA GEMM: HW[n, 0:128] = dropout(relu?(X[n, 0:128])) @ W[128,128]
// Block = 256 threads = 8 waves. Wave w owns output columns [16w, 16w+16).
// Each wave preloads its B (W-slice) fragments into registers (64 VGPRs),
// then iterates 8 row-tiles staged through LDS.
// ---------------------------------------------------------------------------
__global__ __launch_bounds__(256)
void gcn_gemm_dropout(const float* __restrict__ X, const float* __restrict__ W,
                      float* __restrict__ HW, int n_rows, unsigned layer_salt,
                      int do_relu) {
    __shared__ float Xs[16][GCN_D];

    const int tid  = threadIdx.x;
    const int lane = tid & 31;
    const int wave = tid >> 5;          // 0..7 -> 16-column tile
    const int col0 = wave * 16;
    const int half = lane >> 4;         // 0: K+0/1 ; 1: K+2/3  (ISA A/B layout)
    const int l15  = lane & 15;

    // Preload B fragments: B is 4x16 per wmma, lanes0-15 hold N=col, VGPR0/1
    // hold K+2*half, K+2*half+1.   W row-major [K][N].
    v2f bfrag[32];
#pragma unroll
    for (int g = 0; g < 32; ++g) {
        const int k = g * 4 + half * 2;
        bfrag[g].x = W[(k    ) * GCN_D + col0 + l15];
        bfrag[g].y = W[(k + 1) * GCN_D + col0 + l15];
    }

    const long long base_row = (long long)blockIdx.x * GCN_ROWS_PER_WG;

    for (int rt = 0; rt < GCN_ROW_TILES; ++rt) {
        const long long row0 = base_row + rt * 16;

        __syncthreads();   // previous tile fully consumed
        // Stage 16x128 X tile with fused relu + dropout(p=0.5, keep -> x*2)
        for (int i = tid; i < 16 * GCN_D; i += 256) {
            const int r = i >> 7;
            const int c = i & (GCN_D - 1);
            const long long row = row0 + r;
            float v = 0.0f;
            if (row < n_rows) {
                v = X[row * GCN_D + c];
                if (do_relu) v = fmaxf(v, 0.0f);
                const unsigned idx = (unsigned)(row * GCN_D + c);
                const unsigned h = pcg_hash(idx ^ layer_salt);
                v = (h & 1u) ? (v * 2.0f) : 0.0f;
            }
            Xs[r][c] = v;
        }
        __syncthreads();

        v8f acc = {};
#pragma unroll
        for (int g = 0; g < 32; ++g) {
            const int k = g * 4 + half * 2;
            v2f a;
            a.x = Xs[l15][k];
            a.y = Xs[l15][k + 1];
            // 8 args: (neg_a, A, neg_b, B, c_mod, C, reuse_a, reuse_b)
            acc = __builtin_amdgcn_wmma_f32_16x16x4_f32(
                false, a, false, bfrag[g], (short)0, acc, false, false);
        }

        // C/D layout: VGPR r -> M = r (lanes 0-15) or M = r+8 (lanes 16-31),
        // N = lane&15.
#pragma unroll
        for (int r = 0; r < 8; ++r) {
            const long long row = row0 + (half ? (r + 8) : r);
            if (row < n_rows)
                HW[row * GCN_D + col0 + l15] = acc[r];
        }
    }
}

// ---------------------------------------------------------------------------
// agg[n,c] = hw[n,c] * dinv[n]^2 + b[c]   (self-loop + bias; initializes agg)
// ---------------------------------------------------------------------------
__global__ __launch_bounds__(256)
void gcn_self_bias(const float* __restrict__ HW, const float* __restrict__ dinv,
                   const float* __restrict__ b, float* __restrict__ AGG, int n) {
    const long long i = (long long)blockIdx.x * blockDim.x + threadIdx.x;
    const long long total = (long long)n * GCN_D;
    if (i >= total) return;
    const int node = (int)(i >> 7);
    const int c = (int)(i & (GCN_D - 1));
    const float dv = dinv[node];
    AGG[i] = HW[i] * dv * dv + b[c];
}

// ---------------------------------------------------------------------------
// Edge scatter: one wave (32 lanes) per edge; each lane moves 4 consecutive
// floats (global_load_b128 + 4x global_atomic_add_f32).
// ---------------------------------------------------------------------------
__global__ __launch_bounds__(256)
void gcn_edge_scatter(const long long* __restrict__ src,
                      const long long* __restrict__ dst,
                      const float* __restrict__ dinv,
                      const float* __restrict__ HW,
                      float* __restrict__ AGG, int e) {
    const long long t = (long long)blockIdx.x * blockDim.x + threadIdx.x;
    const int lane = (int)(t & 31);
    const long long ed = t >> 5;
    if (ed >= e) return;

    const int s = (int)src[ed];
    const int d = (int)dst[ed];
    const float norm = dinv[s] * dinv[d];

    const float4 v = *(const float4*)(HW + (long long)s * GCN_D + lane * 4);
    float* ad = AGG + (long long)d * GCN_D + lane * 4;
    atomicAdd(ad + 0, v.x * norm);
    atomicAdd(ad + 1, v.y * norm);
    atomicAdd(ad + 2, v.z * norm);
    atomicAdd(ad + 3, v.w * norm);
}

// ---------------------------------------------------------------------------
// Host launcher
// ---------------------------------------------------------------------------
extern "C" void kernel_launch(void* const* d_in, const int* in_sizes, int n_in,
                              void* d_out, int out_size, void* d_ws, size_t ws_size,
                              hipStream_t stream) {
    const float* features       = (const float*)d_in[0];
    const long long* edge_index = (const long long*)d_in[1];
    const float* Wm[3] = { (const float*)d_in[2], (const float*)d_in[4],
                           (const float*)d_in[6] };
    const float* bv[3] = { (const float*)d_in[3], (const float*)d_in[5],
                           (const float*)d_in[7] };

    const int N = in_sizes[0] / GCN_D;        // 100000
    const int E = in_sizes[1] / 2;            // 1600000
    const long long ND = (long long)N * GCN_D;

    const long long* src = edge_index;
    const long long* dst = edge_index + E;

    // Workspace carve: dinv[N] | hw[N*D] | aggA[N*D] | aggB[N*D]
    char* ws = (char*)d_ws;
    size_t off = 0;
    float* dinv = (float*)(ws + off);
    off += ((size_t)N * 4 + 255) & ~(size_t)255;
    float* hw   = (float*)(ws + off); off += (size_t)ND * 4;
    float* aggA = (float*)(ws + off); off += (size_t)ND * 4;
    float* aggB = (float*)(ws + off);
    (void)ws_size;

    // --- degree -> dinv (once) ---
    gcn_zero_f32<<<(N + 255) / 256, 256, 0, stream>>>(dinv, N);
    gcn_deg_count<<<(E + 255) / 256, 256, 0, stream>>>(dst, dinv, E);
    gcn_deg_finalize<<<(N + 255) / 256, 256, 0, stream>>>(dinv, N);

    const int gemm_blocks = (N + GCN_ROWS_PER_WG - 1) / GCN_ROWS_PER_WG;
    const int sb_blocks   = (int)((ND + 255) / 256);
    const int es_blocks   = (int)(((long long)E * 32 + 255) / 256);

    const float* Xin[3]  = { features, aggA, aggB };
    float* aggOut[3]     = { aggA, aggB, (float*)d_out };

    for (int layer = 0; layer < 3; ++layer) {
        const unsigned salt = 0x9E3779B9u * (unsigned)(layer + 1) + 0x85EBCA6Bu;
        gcn_gemm_dropout<<<gemm_blocks, 256, 0, stream>>>(
            Xin[layer], Wm[layer], hw, N, salt, layer > 0 ? 1 : 0);
        gcn_self_bias<<<sb_blocks, 256, 0, stream>>>(
            hw, dinv, bv[layer], aggOut[layer], N);
        gcn_edge_scatter<<<es_blocks, 256, 0, stream>>>(
            src, dst, dinv, hw, aggOut[layer], E);
    }
}